// ROISamplingLayer_66176856097087
// MI455X (gfx1250) — compile-verified
//
#include <hip/hip_runtime.h>
#include <hip/hip_bf16.h>
#include <stdint.h>

// ---------------- problem constants ----------------
#define NB     8
#define NH     64
#define NWID   64
#define NC     256
#define NK     512
#define NCH    7
#define NCW    7
#define NHW    (NH * NWID)                 // 4096
#define NCELLS (NB * NK * NCH * NCW)       // 200704
#define CROPS_ELEMS ((size_t)NCELLS * NC)  // 51380224
#define BOXES_ELEMS (NB * NK * 4)          // 16384

#define HAS_ASYNC_LDS (__has_builtin(__builtin_amdgcn_global_load_async_to_lds_b128))

// =====================================================================
// Kernel 1: per-batch top-512-of-4096 via LDS bitonic sort on u64 keys.
// key = (monotone-mapped score << 32) | (4095 - idx); descending sort
// reproduces jax.lax.top_k ordering (score desc, index asc on ties).
// =====================================================================
__global__ __launch_bounds__(1024) void topk_kernel(
    const float* __restrict__ scores,   // [NB, 4096]
    const float* __restrict__ boxes,    // [NB, 4096, 4]
    float*       __restrict__ out_boxes,// [NB, 512, 4]
    float*       __restrict__ out_idx)  // [NB, 512] (indices as float)
{
  __shared__ unsigned long long key[NHW];   // 32 KB
  const int b   = blockIdx.x;
  const int tid = threadIdx.x;

  for (int i = tid; i < NHW; i += 1024) {
    unsigned u = __float_as_uint(scores[b * NHW + i]);
    u = (u & 0x80000000u) ? ~u : (u | 0x80000000u);   // order-preserving map
    key[i] = ((unsigned long long)u << 32) | (unsigned)(NHW - 1 - i);
  }
  __syncthreads();

  for (int k = 2; k <= NHW; k <<= 1) {
    for (int j = k >> 1; j > 0; j >>= 1) {
      for (int i = tid; i < NHW; i += 1024) {
        int l = i ^ j;
        if (l > i) {
          unsigned long long a = key[i], c = key[l];
          bool desc = ((i & k) == 0);
          if (desc ? (a < c) : (a > c)) { key[i] = c; key[l] = a; }
        }
      }
      __syncthreads();
    }
  }

  if (tid < NK) {
    int idx = (NHW - 1) - (int)(key[tid] & 0xFFFFFFFFu);
    out_idx[b * NK + tid] = (float)idx;
    float4 bx = ((const float4*)boxes)[b * NHW + idx];
    ((float4*)out_boxes)[b * NK + tid] = bx;
  }
}

// =====================================================================
// Kernel 2: crop_and_resize. One wave32 per output cell (7x7 grid pt of
// one box): gather 4 source pixel rows (4 x 1KB), bilinear combine,
// write 1KB. Persistent waves, double-buffered async-to-LDS staging.
// =====================================================================
struct Geom {
  const float* p0; const float* p1; const float* p2; const float* p3;
  float w0, w1, w2, w3;
};

__device__ __forceinline__ Geom make_geom(int cell,
                                          const float* __restrict__ fm,
                                          const float* __restrict__ sboxes) {
  int ix = cell % NCW; int t = cell / NCW;
  int iy = t % NCH;    t /= NCH;
  int n  = t % NK;     int b = t / NK;
  float4 bx = ((const float4*)sboxes)[b * NK + n];   // y1,x1,y2,x2
  float fy = (float)iy * (1.0f / (NCH - 1));
  float fx = (float)ix * (1.0f / (NCW - 1));
  float y = (bx.x + fy * (bx.z - bx.x)) * (float)(NH - 1);
  float x = (bx.y + fx * (bx.w - bx.y)) * (float)(NWID - 1);
  float y0f = floorf(y), x0f = floorf(x);
  float wy = y - y0f,   wx = x - x0f;
  int y0 = min(max((int)y0f, 0), NH - 1);
  int y1i = min(y0 + 1, NH - 1);
  int x0 = min(max((int)x0f, 0), NWID - 1);
  int x1i = min(x0 + 1, NWID - 1);
  const float* base = fm + (size_t)b * NHW * NC;
  Geom g;
  g.p0 = base + (size_t)(y0  * NWID + x0 ) * NC;
  g.p1 = base + (size_t)(y0  * NWID + x1i) * NC;
  g.p2 = base + (size_t)(y1i * NWID + x0 ) * NC;
  g.p3 = base + (size_t)(y1i * NWID + x1i) * NC;
  g.w0 = (1.0f - wy) * (1.0f - wx);
  g.w1 = (1.0f - wy) * wx;
  g.w2 = wy * (1.0f - wx);
  g.w3 = wy * wx;
  return g;
}

__device__ __forceinline__ void fma4(float4& acc, float4 v, float s) {
  acc.x = fmaf(v.x, s, acc.x); acc.y = fmaf(v.y, s, acc.y);
  acc.z = fmaf(v.z, s, acc.z); acc.w = fmaf(v.w, s, acc.w);
}

#if HAS_ASYNC_LDS
// Builtin signature (from clang diagnostic): arg1 = AS1 (global) pointer to
// int __attribute__((vector_size(16))); pass exact types. LDS pointer goes
// through uintptr_t: on AMDGPU the LDS aperture lives in the high 32 bits of
// a generic address, so the low 32 bits ARE the AS3 offset.
typedef int v4i_ __attribute__((vector_size(16)));
typedef __attribute__((address_space(1))) v4i_* as1_v4i;
typedef __attribute__((address_space(3))) v4i_* as3_v4i;

__device__ __forceinline__ void async_cp16(const float* g, float* l) {
  // GLOBAL_LOAD_ASYNC_TO_LDS_B128: 16B/lane, 512B/wave, tracked by ASYNCcnt
  __builtin_amdgcn_global_load_async_to_lds_b128(
      (as1_v4i)(uintptr_t)g,
      (as3_v4i)(uintptr_t)l,
      0, 0);
}
#endif

__device__ __forceinline__ void wait_async_le(int n) {
#if __has_builtin(__builtin_amdgcn_s_wait_asynccnt)
  if (n == 0) __builtin_amdgcn_s_wait_asynccnt(0);
  else        __builtin_amdgcn_s_wait_asynccnt(8);
#else
  if (n == 0) asm volatile("s_wait_asynccnt 0" ::: "memory");
  else        asm volatile("s_wait_asynccnt 8" ::: "memory");
#endif
}

#define WPB 8   // waves per block (256 threads)

__global__ __launch_bounds__(WPB * 32) void crop_kernel(
    const float* __restrict__ fm,      // [NB,NH,NW,NC]
    const float* __restrict__ sboxes,  // [NB,NK,4]
    float*       __restrict__ crops)   // [NB,NK,NCH,NCW,NC]
{
  const int lane = threadIdx.x & 31;
  const int wib  = threadIdx.x >> 5;
  const int gw   = blockIdx.x * WPB + wib;
  const int nw   = gridDim.x * WPB;

#if HAS_ASYNC_LDS
  __shared__ float stage[WPB][2][4][NC];   // 64 KB: 2 bufs x 4 rows x 1KB per wave

  auto issue = [&](int cell, int buf) {
    Geom g = make_geom(cell, fm, sboxes);
    const float* ps[4] = {g.p0, g.p1, g.p2, g.p3};
#pragma unroll
    for (int r = 0; r < 4; ++r) {
      async_cp16(ps[r] + lane * 4,       &stage[wib][buf][r][lane * 4]);
      async_cp16(ps[r] + 128 + lane * 4, &stage[wib][buf][r][128 + lane * 4]);
    }
  };

  auto consume = [&](int cell, int buf) {
    Geom g = make_geom(cell, fm, sboxes);
    const float w[4] = {g.w0, g.w1, g.w2, g.w3};
    const int c0 = lane * 8;
    float4 acc0 = make_float4(0.f, 0.f, 0.f, 0.f);
    float4 acc1 = make_float4(0.f, 0.f, 0.f, 0.f);
#pragma unroll
    for (int r = 0; r < 4; ++r) {
      float4 a = *(const float4*)&stage[wib][buf][r][c0];
      float4 c = *(const float4*)&stage[wib][buf][r][c0 + 4];
      fma4(acc0, a, w[r]);
      fma4(acc1, c, w[r]);
    }
    float* o = crops + (size_t)cell * NC + c0;
    *(float4*)o       = acc0;
    *(float4*)(o + 4) = acc1;
  };

  int cell = gw;
  if (cell < NCELLS) issue(cell, 0);
  int buf = 0;
  for (; cell < NCELLS; cell += nw) {
    int nxt = cell + nw;
    if (nxt < NCELLS) {               // prefetch next cell into other buffer
      issue(nxt, buf ^ 1);
      asm volatile("" ::: "memory");
      wait_async_le(8);               // oldest 8 (current buf) complete
    } else {
      asm volatile("" ::: "memory");
      wait_async_le(0);
    }
    asm volatile("" ::: "memory");
    consume(cell, buf);
    buf ^= 1;
  }
#else
  // Fallback: direct vectorized gather (b128 loads) through L2-resident fm.
  for (int cell = gw; cell < NCELLS; cell += nw) {
    Geom g = make_geom(cell, fm, sboxes);
    const float* ps[4] = {g.p0, g.p1, g.p2, g.p3};
    const float w[4] = {g.w0, g.w1, g.w2, g.w3};
    const int c0 = lane * 8;
    float4 acc0 = make_float4(0.f, 0.f, 0.f, 0.f);
    float4 acc1 = make_float4(0.f, 0.f, 0.f, 0.f);
#pragma unroll
    for (int r = 0; r < 4; ++r) {
      float4 a = *(const float4*)(ps[r] + c0);
      float4 c = *(const float4*)(ps[r] + c0 + 4);
      fma4(acc0, a, w[r]);
      fma4(acc1, c, w[r]);
    }
    float* o = crops + (size_t)cell * NC + c0;
    *(float4*)o       = acc0;
    *(float4*)(o + 4) = acc1;
  }
#endif
}

// =====================================================================
extern "C" void kernel_launch(void* const* d_in, const int* in_sizes, int n_in,
                              void* d_out, int out_size, void* d_ws, size_t ws_size,
                              hipStream_t stream) {
  (void)in_sizes; (void)n_in; (void)out_size; (void)d_ws; (void)ws_size;
  const float* fm     = (const float*)d_in[0];  // [8,64,64,256]
  const float* boxes  = (const float*)d_in[1];  // [8,64,64,4]
  const float* scores = (const float*)d_in[2];  // [8,64,64,1]

  float* crops     = (float*)d_out;             // [8,512,7,7,256]
  float* out_boxes = crops + CROPS_ELEMS;       // [8,512,4]
  float* out_idx   = out_boxes + BOXES_ELEMS;   // [8,512] as float

  topk_kernel<<<NB, 1024, 0, stream>>>(scores, boxes, out_boxes, out_idx);
  crop_kernel<<<1024, WPB * 32, 0, stream>>>(fm, out_boxes, crops);
}